// TrainGRPOModulePallas_8589934595
// MI455X (gfx1250) — compile-verified
//
#include <hip/hip_runtime.h>
#include <cstdint>

#define LOG2E 1.4426950408889634f
#define LN2   0.6931471805599453f
#define WAVES 8
#define DEPTH 4

__device__ __forceinline__ float fast_exp2(float x) {
#if __has_builtin(__builtin_amdgcn_exp2f)
  return __builtin_amdgcn_exp2f(x);   // v_exp_f32
#else
  return exp2f(x);
#endif
}
__device__ __forceinline__ float fast_log2(float x) {
#if __has_builtin(__builtin_amdgcn_logf)
  return __builtin_amdgcn_logf(x);    // v_log_f32 (base-2)
#else
  return log2f(x);
#endif
}

// ---------------------------------------------------------------------------
// Kernel 1: one block per (b,t) row (t < T-1), grid = (T-1, B).
// Single pass over V logits streamed HBM -> LDS with gfx1250 async-DMA
// (depth-4 per-wave ring, non-temporal), computing online base-2 softmax
// stats + chosen-logit gather.
// ---------------------------------------------------------------------------
__global__ __launch_bounds__(256) void grpo_row_stats(
    const float* __restrict__ logits,
    const int*   __restrict__ input_ids,
    float* __restrict__ ws_plp,
    float* __restrict__ ws_ent,
    int T, int V)
{
  const int Tm1 = T - 1;
  const int t   = blockIdx.x;           // 0 .. T-2
  const int b   = blockIdx.y;           // 0 .. B-1
  const int row = b * Tm1 + t;
  const float* rowp = logits + (long long)(b * T + t) * (long long)V;
  const int chosen = input_ids[b * T + t + 1];
  const int c4   = chosen >> 2;
  const int csub = chosen & 3;

  const int tid  = threadIdx.x;
  const int lane = tid & 31;
  const int w    = tid >> 5;            // wave id, 0..7 (wave32)

  const int NCH = V >> 7;               // 512B wave-chunks per row (V % 128 == 0)
  const int n   = (NCH - w + WAVES - 1) / WAVES;   // chunks owned by this wave

  __shared__ float4 smem[DEPTH * WAVES * 32];      // 16 KB ring buffers
  __shared__ float4 red[WAVES];

  float m  = -__builtin_inff();
  float s  = 0.0f;                      // sum 2^(y - m), y = x*log2e
  float u  = 0.0f;                      // sum x * 2^(y - m)
  float xc = -__builtin_inff();         // chosen logit (one lane finds it)

  auto issue = [&](int k) {
    const int c = k * WAVES + w;        // global chunk id
    const float4* g = reinterpret_cast<const float4*>(rowp) + (c * 32 + lane);
    unsigned laddr = (unsigned)(uintptr_t)
        (&smem[((k & (DEPTH - 1)) * WAVES + w) * 32 + lane]);
    unsigned long long ga = (unsigned long long)(uintptr_t)g;
    // Wait for our pending LDS reads before overwriting the ring slot, then
    // fire the async 16B/lane (512B/wave) non-temporal DMA into LDS.
    asm volatile("s_wait_dscnt 0x0\n\t"
                 "global_load_async_to_lds_b128 %0, %1, off th:TH_LOAD_NT"
                 :: "v"(laddr), "v"(ga) : "memory");
  };

  auto consume = [&](int k) {
    const float4 q = smem[((k & (DEPTH - 1)) * WAVES + w) * 32 + lane];
    const int f = (k * WAVES + w) * 32 + lane;     // float4 index in row
    const float x0 = q.x, x1 = q.y, x2 = q.z, x3 = q.w;
    const float y0 = x0 * LOG2E, y1 = x1 * LOG2E;
    const float y2 = x2 * LOG2E, y3 = x3 * LOG2E;
    const float m4 = fmaxf(fmaxf(y0, y1), fmaxf(y2, y3));
    const float mn = fmaxf(m, m4);
    const float sc = fast_exp2(m - mn);            // exp2(-inf)=0 on first use
    const float e0 = fast_exp2(y0 - mn), e1 = fast_exp2(y1 - mn);
    const float e2 = fast_exp2(y2 - mn), e3 = fast_exp2(y3 - mn);
    s = fmaf(s, sc, (e0 + e1) + (e2 + e3));
    u = fmaf(u, sc, fmaf(x0, e0, fmaf(x1, e1, fmaf(x2, e2, x3 * e3))));
    m = mn;
    if (f == c4) {
      float v = q.x;
      v = (csub == 1) ? q.y : v;
      v = (csub == 2) ? q.z : v;
      v = (csub == 3) ? q.w : v;
      xc = v;
    }
  };

  // -------- depth-4 async pipeline --------
  const int pre = n < DEPTH ? n : DEPTH;
  for (int k = 0; k < pre; ++k) issue(k);
  const int steady_end = n - DEPTH;                // may be <= 0
  for (int k = 0; k < steady_end; ++k) {
    asm volatile("s_wait_asynccnt 0x3" ::: "memory");   // oldest chunk landed
    consume(k);
    issue(k + DEPTH);
  }
  asm volatile("s_wait_asynccnt 0x0" ::: "memory");
  for (int k = (steady_end > 0 ? steady_end : 0); k < n; ++k) consume(k);

  // -------- wave32 reduction --------
  for (int off = 16; off >= 1; off >>= 1) {
    const float m2 = __shfl_xor(m,  off, 32);
    const float s2 = __shfl_xor(s,  off, 32);
    const float u2 = __shfl_xor(u,  off, 32);
    const float x2 = __shfl_xor(xc, off, 32);
    const float mn = fmaxf(m, m2);
    const float a  = fast_exp2(m  - mn);
    const float bb = fast_exp2(m2 - mn);
    s  = s * a + s2 * bb;
    u  = u * a + u2 * bb;
    m  = mn;
    xc = fmaxf(xc, x2);
  }
  if (lane == 0) red[w] = make_float4(m, s, u, xc);
  __syncthreads();

  if (tid == 0) {
    float mm = red[0].x, ss = red[0].y, uu = red[0].z, xx = red[0].w;
    for (int i = 1; i < WAVES; ++i) {
      const float4 r = red[i];
      const float mn = fmaxf(mm, r.x);
      const float a  = fast_exp2(mm  - mn);
      const float bb = fast_exp2(r.x - mn);
      ss = ss * a + r.y * bb;
      uu = uu * a + r.z * bb;
      xx = fmaxf(xx, r.w);
      mm = mn;
    }
    const float logZ = LN2 * (mm + fast_log2(ss));  // natural log partition
    ws_ent[row] = logZ - uu / ss;                   // -sum p*logp
    ws_plp[row] = xx - logZ;                        // logp of chosen token
  }
}

// ---------------------------------------------------------------------------
// Kernel 2: per-batch finalize (T-1 tokens). Deterministic serial scans.
// ---------------------------------------------------------------------------
__global__ __launch_bounds__(256) void grpo_finalize(
    const float* __restrict__ ws_plp,
    const float* __restrict__ ws_ent,
    const float* __restrict__ old_lps,
    const float* __restrict__ advantages,
    const float* __restrict__ labels,
    float* __restrict__ out,
    float* __restrict__ ws_part,
    int T, int B)
{
  const int b   = blockIdx.x;
  const int tid = threadIdx.x;
  const int Tm1 = T - 1;
  const float adv = advantages[b];

  __shared__ int   cnt[256];
  __shared__ int   excl[256];
  __shared__ float rbuf[256];

  float lsum = 0.f, msum = 0.f, esum = 0.f;
  float entl[4]; bool vall[4]; int nval = 0;

  const int t0 = tid * 4;
  for (int j = 0; j < 4; ++j) {
    const int t = t0 + j;
    entl[j] = 0.f; vall[j] = false;
    if (t < Tm1) {
      const float plp  = ws_plp[b * Tm1 + t];
      const float ent  = ws_ent[b * Tm1 + t];
      const float mask = labels[b * T + t + 1];
      out[1 + b * Tm1 + t] = plp;                       // per_token_logps_out
      const float ratio = fast_exp2((plp - old_lps[b * Tm1 + t]) * LOG2E);
      const float unc = ratio * adv;
      const float cl  = fminf(fmaxf(ratio, 0.8f), 1.3f) * adv;
      lsum += (-fminf(unc, cl)) * mask;
      msum += mask;
      esum += ent * mask;
      entl[j] = ent;
      vall[j] = (mask == 1.0f);
      if (vall[j]) ++nval;
    }
  }
  cnt[tid] = nval;
  __syncthreads();
  if (tid == 0) {                    // tiny exclusive scan over 256 entries
    int run = 0;
    for (int i = 0; i < 256; ++i) { excl[i] = run; run += cnt[i]; }
  }
  __syncthreads();

  int cum = excl[tid];
  float tsum = 0.f, tcnt = 0.f;
  for (int j = 0; j < 4; ++j) {
    if (vall[j]) {
      ++cum;
      if (cum >= 4 && cum <= 100) { tsum += entl[j]; tcnt += 1.f; }
    }
  }

  float sm_total = 0.f, ts_total = 0.f;   // live only in thread 0
  rbuf[tid] = lsum; __syncthreads();
  if (tid == 0) { float a = 0.f; for (int i = 0; i < 256; ++i) a += rbuf[i]; ws_part[2 * b + 0] = a; }
  __syncthreads();
  rbuf[tid] = msum; __syncthreads();
  if (tid == 0) { float a = 0.f; for (int i = 0; i < 256; ++i) a += rbuf[i]; sm_total = a; ws_part[2 * b + 1] = a; }
  __syncthreads();
  rbuf[tid] = esum; __syncthreads();
  if (tid == 0) { float a = 0.f; for (int i = 0; i < 256; ++i) a += rbuf[i]; out[1 + B * Tm1 + b] = a / sm_total; }
  __syncthreads();
  rbuf[tid] = tsum; __syncthreads();
  if (tid == 0) { float a = 0.f; for (int i = 0; i < 256; ++i) a += rbuf[i]; ts_total = a; }
  __syncthreads();
  rbuf[tid] = tcnt; __syncthreads();
  if (tid == 0) { float a = 0.f; for (int i = 0; i < 256; ++i) a += rbuf[i]; out[1 + B * Tm1 + B + b] = ts_total / a; }
}

// ---------------------------------------------------------------------------
// Kernel 3: combine per-batch loss partials with global normalization.
// ---------------------------------------------------------------------------
__global__ void grpo_loss(const float* __restrict__ ws_part,
                          float* __restrict__ out, int B)
{
  float l = 0.f, mm = 0.f;
  for (int b = 0; b < B; ++b) { l += ws_part[2 * b]; mm += ws_part[2 * b + 1]; }
  out[0] = l / mm;
}

extern "C" void kernel_launch(void* const* d_in, const int* in_sizes, int n_in,
                              void* d_out, int out_size, void* d_ws, size_t ws_size,
                              hipStream_t stream) {
  const float* logits     = (const float*)d_in[0];
  const int*   input_ids  = (const int*)  d_in[1];
  const float* old_lps    = (const float*)d_in[2];
  const float* advantages = (const float*)d_in[3];
  const float* labels     = (const float*)d_in[4];
  float* out = (float*)d_out;

  const int BT = in_sizes[1];          // B*T
  const int B  = in_sizes[3];
  const int T  = BT / B;
  const int V  = in_sizes[0] / BT;
  const int Tm1  = T - 1;
  const int rows = B * Tm1;

  float* ws_plp  = (float*)d_ws;       // rows
  float* ws_ent  = ws_plp + rows;      // rows
  float* ws_part = ws_ent + rows;      // 2*B

  dim3 grid1(Tm1, B, 1);
  grpo_row_stats<<<grid1, 256, 0, stream>>>(logits, input_ids, ws_plp, ws_ent, T, V);
  grpo_finalize<<<B, 256, 0, stream>>>(ws_plp, ws_ent, old_lps, advantages, labels,
                                       out, ws_part, T, B);
  grpo_loss<<<1, 1, 0, stream>>>(ws_part, out, B);
}